// DiagWeightLayer_7765300871196
// MI455X (gfx1250) — compile-verified
//
#include <hip/hip_runtime.h>
#include <hip/hip_bf16.h>

// out[r, c] = x[r, c] * weight[c], x: 32768x1024 f32, weight: 1024 f32.
// Pure streaming op: ~268 MB traffic @ 23.3 TB/s HBM => ~11.5 us floor.
// b128 non-temporal loads/stores for the two 128 MiB dead streams (combined
// they exceed the 192 MB L2), RT-cached weight load (4 KB, stays in WGP$),
// 4 float4s in flight per thread for memory-level parallelism.

typedef float v4f __attribute__((ext_vector_type(4)));

constexpr int TPB = 256;  // 8 wave32 waves per workgroup
constexpr int VPT = 4;    // float4 elements per thread (MLP depth)

// Preconditions (checked on host): cols4 is a power of two and the grid
// stride (gridDim.x * TPB) is a multiple of cols4, so each thread's weight
// vector is loop-invariant: (i + k*stride) & wmask == i & wmask.
__global__ __launch_bounds__(TPB) void diag_scale_vec(
    const v4f* __restrict__ x, const v4f* __restrict__ w, v4f* __restrict__ out,
    unsigned n4, unsigned wmask) {
  const unsigned stride = gridDim.x * TPB;
  unsigned i = blockIdx.x * TPB + threadIdx.x;

  const v4f wv = w[i & wmask];  // regular (RT) cached load, loop-invariant

  // 4 streamed float4s per iteration; loads clustered for 4 outstanding
  // loadcnt slots before the first wait.
  while (i + 3u * stride < n4) {
    v4f a0 = __builtin_nontemporal_load(x + i);
    v4f a1 = __builtin_nontemporal_load(x + i + stride);
    v4f a2 = __builtin_nontemporal_load(x + i + 2u * stride);
    v4f a3 = __builtin_nontemporal_load(x + i + 3u * stride);
    __builtin_nontemporal_store(a0 * wv, out + i);
    __builtin_nontemporal_store(a1 * wv, out + i + stride);
    __builtin_nontemporal_store(a2 * wv, out + i + 2u * stride);
    __builtin_nontemporal_store(a3 * wv, out + i + 3u * stride);
    i += 4u * stride;
  }
  // Tail (empty for the harness shape: 8,388,608 float4s tile exactly).
  for (; i < n4; i += stride) {
    v4f a = __builtin_nontemporal_load(x + i);
    __builtin_nontemporal_store(a * wv, out + i);
  }
}

// Minimal fallback for shapes that violate the vector-path preconditions
// (never hit by the harness: 32768x1024 f32).
__global__ __launch_bounds__(TPB) void diag_scale_scalar(
    const float* __restrict__ x, const float* __restrict__ w,
    float* __restrict__ out, unsigned n, unsigned cols) {
  const unsigned stride = gridDim.x * TPB;
  for (unsigned i = blockIdx.x * TPB + threadIdx.x; i < n; i += stride) {
    out[i] = x[i] * w[i % cols];
  }
}

extern "C" void kernel_launch(void* const* d_in, const int* in_sizes, int n_in,
                              void* d_out, int out_size, void* d_ws, size_t ws_size,
                              hipStream_t stream) {
  const float* x = (const float*)d_in[0];
  const float* w = (const float*)d_in[1];
  float* out = (float*)d_out;

  const unsigned n = (unsigned)in_sizes[0];     // 33,554,432 floats
  const unsigned cols = (unsigned)in_sizes[1];  // 1024

  const unsigned n4 = n / 4u;
  const unsigned cols4 = cols / 4u;  // 256 for harness shape

  unsigned blocks = (n4 + (unsigned)(TPB * VPT) - 1u) / (unsigned)(TPB * VPT);
  if (blocks == 0u) blocks = 1u;
  const unsigned long long stride = (unsigned long long)blocks * (unsigned)TPB;

  const bool pow2 = cols4 != 0u && (cols4 & (cols4 - 1u)) == 0u;
  const bool vec_ok =
      (n % 4u) == 0u && (cols % 4u) == 0u && pow2 && (stride % cols4) == 0ull;

  if (vec_ok) {
    // Harness shape: 8192 blocks x 256 threads, each thread 4 float4s, exact
    // tiling, weight vector = w4[threadIdx.x & 255] hoisted out of the loop.
    diag_scale_vec<<<blocks, TPB, 0, stream>>>(
        (const v4f*)x, (const v4f*)w, (v4f*)out, n4, cols4 - 1u);
  } else {
    unsigned sblocks = (n + TPB - 1u) / TPB;
    if (sblocks > 65535u * 8u) sblocks = 65535u * 8u;
    if (sblocks == 0u) sblocks = 1u;
    diag_scale_scalar<<<sblocks, TPB, 0, stream>>>(x, w, out, n, cols);
  }
}